// ExampleTiedDropout_48129403519286
// MI455X (gfx1250) — compile-verified
//
#include <hip/hip_runtime.h>

// ExampleTiedDropout on MI455X (gfx1250, wave32).
// Bandwidth-bound elementwise: 268 MB traffic @ 23.3 TB/s => ~11.5 us floor.
// Strategy: exact JAX Threefry-2x32 mask generation into d_ws (tiny), then a
// b128 non-temporal streaming multiply with next-plane global_prefetch.

typedef unsigned int u32;
typedef float v4f __attribute__((ext_vector_type(4)));

#define FIXED_CH 51   // int(0.2 * 256)
#define MEM_CH   205  // 256 - 51
#define PAIRS    103  // ceil(205 / 2) counter pairs after padding
#define NCH      256
#define PLANE    1024 // 32*32
#define NPLANES  (128 * 256)

__device__ __forceinline__ u32 rotl32(u32 x, u32 d) {
  return (x << d) | (x >> (32u - d));
}

// JAX threefry2x32: 20 rounds, key schedule every 4 rounds.
__device__ __forceinline__ void threefry2x32(u32 k0, u32 k1, u32 x0, u32 x1,
                                             u32& o0, u32& o1) {
  const u32 ks0 = k0, ks1 = k1, ks2 = k0 ^ k1 ^ 0x1BD11BDAu;
  x0 += ks0;
  x1 += ks1;
  const u32 rotA[4] = {13u, 15u, 26u, 6u};
  const u32 rotB[4] = {17u, 29u, 16u, 24u};
  const u32 ks[3] = {ks0, ks1, ks2};
#pragma unroll
  for (int i = 0; i < 5; ++i) {
    const u32* r = (i & 1) ? rotB : rotA;
#pragma unroll
    for (int j = 0; j < 4; ++j) {
      x0 += x1;
      x1 = rotl32(x1, r[j]);
      x1 ^= x0;
    }
    x0 += ks[(i + 1) % 3];
    x1 += ks[(i + 2) % 3] + (u32)(i + 1);
  }
  o0 = x0;
  o1 = x1;
}

__device__ __forceinline__ float bits_to_mask(u32 bits) {
  // uniform [0,1): bitcast((bits>>9)|0x3f800000) - 1.0 ; bernoulli: u < 0.1
  float u = __uint_as_float((bits >> 9) | 0x3F800000u) - 1.0f;
  return (u < 0.1f) ? 1.0f : 0.0f;
}

// One block per example. Thread t in [0,103) computes Threefry pair
// (t, 103+t) (pair 102 is (102, 0) from JAX's odd-size zero pad); out0 covers
// mem channels [0,103), out1 covers [103,205) (out1 of pair 102 is dropped).
__global__ void tied_dropout_masks(const int* __restrict__ indices,
                                   float* __restrict__ masks) {
  const int b = blockIdx.x;
  const int t = threadIdx.x;  // 0..127
  const u32 idx = (u32)indices[b];

  // fold_in: key = threefry2x32(key(42)=(0,42), seed(idx)=(0,idx))
  u32 k0, k1;
  threefry2x32(0u, 42u, 0u, idx, k0, k1);

  float* m = masks + (size_t)b * NCH;
  if (t < FIXED_CH) m[t] = 1.0f;  // generalization channels always on

  if (t < PAIRS) {
    const u32 x0 = (u32)t;
    const u32 x1 = (t < PAIRS - 1) ? (u32)(PAIRS + t) : 0u;  // zero pad
    u32 o0, o1;
    threefry2x32(k0, k1, x0, x1, o0, o1);
    m[FIXED_CH + t] = bits_to_mask(o0);
    if (t < PAIRS - 1) {
      m[FIXED_CH + PAIRS + t] = bits_to_mask(o1);
    }
  }
}

// One block per (b,c) plane: 256 threads x float4 = 1024 floats.
// Non-temporal b128 load/store (touch-once streams) + prefetch of next plane.
__global__ void __launch_bounds__(256)
tied_dropout_scale(const float* __restrict__ X,
                   const float* __restrict__ masks,
                   float* __restrict__ out) {
  const int plane = blockIdx.x;   // b*C + c
  const int t = threadIdx.x;      // 0..255
  const float m = masks[plane];
  const size_t base = (size_t)plane * PLANE + (size_t)t * 4;

  // Hint the next plane's cachelines into the hierarchy (gfx1250
  // global_prefetch_b8; no counter, silently dropped if invalid).
  if (plane + 1 < NPLANES) {
    __builtin_prefetch(X + base + PLANE, 0, 1);
  }

  v4f v = __builtin_nontemporal_load((const v4f*)(X + base));
  v.x *= m;
  v.y *= m;
  v.z *= m;
  v.w *= m;
  __builtin_nontemporal_store(v, (v4f*)(out + base));
}

extern "C" void kernel_launch(void* const* d_in, const int* in_sizes, int n_in,
                              void* d_out, int out_size, void* d_ws, size_t ws_size,
                              hipStream_t stream) {
  const float* X = (const float*)d_in[0];
  const int* indices = (const int*)d_in[1];
  float* out = (float*)d_out;
  float* masks = (float*)d_ws;  // 128 * 256 floats = 128 KB scratch

  // 1) Per-example Threefry channel masks (tiny: ~13K block ciphers).
  tied_dropout_masks<<<dim3(128), dim3(128), 0, stream>>>(indices, masks);

  // 2) Streaming multiply: 32768 planes x 1024 floats, b128 NT in/out.
  tied_dropout_scale<<<dim3(NPLANES), dim3(256), 0, stream>>>(X, masks, out);
}